// MultimodalLlamaModelMultiTokens_1133871366502
// MI455X (gfx1250) — compile-verified
//
#include <hip/hip_runtime.h>
#include <math.h>
#include <stdint.h>

constexpr int cB = 4, cS = 1024, cD = 2048, cNH = 16, cNKV = 8, cHD = 128;
constexpr int cFFN = 5632, cV = 32000, cLAT = 1024, cHID = 2048, cK = 8;
constexpr float cEPS = 1e-5f;

typedef float v2f __attribute__((ext_vector_type(2)));
typedef float v8f __attribute__((ext_vector_type(8)));

__device__ __forceinline__ float wave_reduce_sum(float s) {
#pragma unroll
  for (int off = 16; off > 0; off >>= 1) s += __shfl_xor(s, off, 32);
  return s;
}

// Async global->LDS 128b copy (CDNA5 GLOBAL_LOAD_ASYNC_TO_LDS_B128, ASYNCcnt).
// lds_off: per-lane LDS byte offset; gptr: per-lane 64-bit global address.
__device__ __forceinline__ void async_copy_b128(uint32_t lds_off, const float* gptr) {
  asm volatile("global_load_async_to_lds_b128 %0, %1, off"
               :: "v"(lds_off), "v"((uint64_t)(uintptr_t)gptr)
               : "memory");
}
__device__ __forceinline__ void wait_async_le4() {
  asm volatile("s_wait_asynccnt 0x4" ::: "memory");
}
__device__ __forceinline__ void wait_async_0() {
  asm volatile("s_wait_asynccnt 0x0" ::: "memory");
}

// ---------------------------------------------------------------------------
// fp32 WMMA GEMM: C[M,N] (+)= A[M,K] * W[K,N]
// Block: 256 threads = 8 waves; block tile 64x64; wave tile 32x16
// (two v8f accumulators sharing one B fragment -> 16 wmma per 32-K chunk).
// Tiles double-buffered in LDS, filled by async global->LDS B128 copies;
// consumer syncs via s_wait_asynccnt + workgroup barrier.
// Fragment layouts per CDNA5 ISA 7.12.2:
//   A 16x4 : lanes 0-15 M=lane,K={0,1}; lanes 16-31 M=lane-16,K={2,3}
//   B 4x16 : VGPR v, lanes 0-15 K=v,N=lane; lanes 16-31 K=v+2,N=lane-16
//   C 16x16: VGPR v, lanes 0-15 M=v,N=lane; lanes 16-31 M=v+8,N=lane-16
// ---------------------------------------------------------------------------
__global__ __launch_bounds__(256) void gemm_wmma_f32(
    const float* __restrict__ A, const float* __restrict__ W,
    float* __restrict__ C, int M, int N, int Kd, int residual)
{
  __shared__ float As[2][64][36];   // 64x32 A tile, padded rows (16B aligned, bank-clean)
  __shared__ float Bs[2][32][68];   // 32x64 B tile, padded rows

  const int t    = threadIdx.x;
  const int lane = t & 31;
  const int wvid = t >> 5;            // 0..7
  const int wm   = wvid >> 2;         // 0..1 (32-row slab)
  const int wn   = wvid & 3;          // 0..3 (16-col slab)
  const int half = lane >> 4;         // 0/1
  const int l16  = lane & 15;
  const int arow0 = wm * 32 + l16;    // first 16x16 tile rows
  const int arow1 = arow0 + 16;       // second 16x16 tile rows
  const int koff  = half * 2;
  const int bcol  = wn * 16 + l16;

  const int m0 = blockIdx.y * 64;
  const int n0 = blockIdx.x * 64;

  // cooperative async-load geometry (256 threads, 4 x B128 per thread per tile)
  const int a_r0 = t >> 3;            // 0..31
  const int a_r1 = a_r0 + 32;         // 32..63
  const int a_c  = (t & 7) << 2;      // 0..28
  const int b_r0 = t >> 4;            // 0..15
  const int b_c  = (t & 15) << 2;     // 0..60

  // per-thread LDS destinations (byte offsets), both buffers
  uint32_t ldsA0[2], ldsA1[2], ldsB0[2], ldsB1[2];
#pragma unroll
  for (int bf = 0; bf < 2; ++bf) {
    ldsA0[bf] = (uint32_t)(uintptr_t)&As[bf][a_r0][a_c];
    ldsA1[bf] = (uint32_t)(uintptr_t)&As[bf][a_r1][a_c];
    ldsB0[bf] = (uint32_t)(uintptr_t)&Bs[bf][b_r0][b_c];
    ldsB1[bf] = (uint32_t)(uintptr_t)&Bs[bf][b_r0 + 16][b_c];
  }

  const float* gA0 = &A[(size_t)(m0 + a_r0) * Kd + a_c];
  const float* gA1 = &A[(size_t)(m0 + a_r1) * Kd + a_c];
  const float* gB0 = &W[(size_t)b_r0 * N + n0 + b_c];
  const float* gB1 = &W[(size_t)(b_r0 + 16) * N + n0 + b_c];

  v8f acc0 = {0.f, 0.f, 0.f, 0.f, 0.f, 0.f, 0.f, 0.f};
  v8f acc1 = {0.f, 0.f, 0.f, 0.f, 0.f, 0.f, 0.f, 0.f};

  const int T = Kd >> 5;  // K chunks of 32

  // prologue: stage tile 0 into buffer 0
  async_copy_b128(ldsA0[0], gA0);
  async_copy_b128(ldsA1[0], gA1);
  async_copy_b128(ldsB0[0], gB0);
  async_copy_b128(ldsB1[0], gB1);

  int cb = 0;
  for (int it = 0; it < T; ++it) {
    const bool more = (it + 1) < T;
    if (more) {  // stage next tile into the other buffer while we compute
      const size_t k1 = (size_t)(it + 1) << 5;
      const int nb = cb ^ 1;
      async_copy_b128(ldsA0[nb], gA0 + k1);
      async_copy_b128(ldsA1[nb], gA1 + k1);
      async_copy_b128(ldsB0[nb], gB0 + k1 * N);
      async_copy_b128(ldsB1[nb], gB1 + k1 * N);
      wait_async_le4();   // current tile's 4 copies done; next 4 still in flight
    } else {
      wait_async_0();
    }
    __syncthreads();
#pragma unroll
    for (int kk = 0; kk < 32; kk += 4) {
      v2f b;
      b.x = Bs[cb][kk + koff][bcol];
      b.y = Bs[cb][kk + koff + 1][bcol];
      v2f a0 = *(const v2f*)&As[cb][arow0][kk + koff];
      v2f a1 = *(const v2f*)&As[cb][arow1][kk + koff];
      acc0 = __builtin_amdgcn_wmma_f32_16x16x4_f32(false, a0, false, b,
                                                   (short)0, acc0, false, false);
      acc1 = __builtin_amdgcn_wmma_f32_16x16x4_f32(false, a1, false, b,
                                                   (short)0, acc1, false, false);
    }
    __syncthreads();
    cb ^= 1;
  }

  const int crow0 = m0 + wm * 32 + half * 8;
  const int ccol  = n0 + bcol;
#pragma unroll
  for (int vv = 0; vv < 8; ++vv) {
    size_t i0 = (size_t)(crow0 + vv) * N + ccol;
    size_t i1 = (size_t)(crow0 + 16 + vv) * N + ccol;
    if (residual) { C[i0] += acc0[vv]; C[i1] += acc1[vv]; }
    else          { C[i0]  = acc0[vv]; C[i1]  = acc1[vv]; }
  }
}

// ---------------------------------------------------------------------------
// RMSNorm: one 256-thread block per row of D=2048
// ---------------------------------------------------------------------------
__global__ __launch_bounds__(256) void rmsnorm_kernel(
    const float* __restrict__ in, const float* __restrict__ w, float* __restrict__ out)
{
  __shared__ float red[8];
  const int r = blockIdx.x, t = threadIdx.x;
  const float* x = in + (size_t)r * cD;
  float s = 0.f;
#pragma unroll
  for (int j = 0; j < 8; ++j) { float u = x[t + 256 * j]; s += u * u; }
  s = wave_reduce_sum(s);
  if ((t & 31) == 0) red[t >> 5] = s;
  __syncthreads();
  float tot = 0.f;
#pragma unroll
  for (int j = 0; j < 8; ++j) tot += red[j];
  const float rs = rsqrtf(tot / (float)cD + cEPS);
#pragma unroll
  for (int j = 0; j < 8; ++j) {
    int c = t + 256 * j;
    out[(size_t)r * cD + c] = x[c] * rs * w[c];
  }
}

// ---------------------------------------------------------------------------
// LayerNorm (in-place) for spectral tokens: B*K rows of D
// ---------------------------------------------------------------------------
__global__ __launch_bounds__(256) void layernorm_kernel(
    float* __restrict__ xio, const float* __restrict__ g, const float* __restrict__ bb)
{
  __shared__ float red1[8], red2[8];
  const int r = blockIdx.x, t = threadIdx.x;
  float* x = xio + (size_t)r * cD;
  float s1 = 0.f, s2 = 0.f;
#pragma unroll
  for (int j = 0; j < 8; ++j) { float u = x[t + 256 * j]; s1 += u; s2 += u * u; }
  s1 = wave_reduce_sum(s1);
  s2 = wave_reduce_sum(s2);
  if ((t & 31) == 0) { red1[t >> 5] = s1; red2[t >> 5] = s2; }
  __syncthreads();
  float S1 = 0.f, S2 = 0.f;
#pragma unroll
  for (int j = 0; j < 8; ++j) { S1 += red1[j]; S2 += red2[j]; }
  const float mu  = S1 / (float)cD;
  const float var = S2 / (float)cD - mu * mu;
  const float inv = rsqrtf(var + cEPS);
#pragma unroll
  for (int j = 0; j < 8; ++j) {
    int c = t + 256 * j;
    x[c] = (x[c] - mu) * inv * g[c] + bb[c];
  }
}

// ---------------------------------------------------------------------------
// Spectra MLP stage 1: (B,LAT) @ (LAT,HID) + b, exact GELU. Wave per output.
// ---------------------------------------------------------------------------
__global__ __launch_bounds__(256) void mlp1_kernel(
    const float* __restrict__ sp, const float* __restrict__ w,
    const float* __restrict__ bias, float* __restrict__ out)
{
  const int wid = blockIdx.x * 8 + (threadIdx.x >> 5);
  const int lane = threadIdx.x & 31;
  const int o = wid & (cHID - 1);
  const int b = wid >> 11;
  float s = 0.f;
#pragma unroll 4
  for (int j = 0; j < cLAT / 32; ++j) {
    int kx = lane + 32 * j;
    s += sp[b * cLAT + kx] * w[(size_t)kx * cHID + o];
  }
  s = wave_reduce_sum(s);
  if (lane == 0) {
    float y = s + bias[o];
    out[b * cHID + o] = 0.5f * y * (1.f + erff(y * 0.7071067811865475f));
  }
}

// ---------------------------------------------------------------------------
// Spectra MLP stage 2: (B,HID) @ (HID,K*D) + b. Wave per output.
// ---------------------------------------------------------------------------
__global__ __launch_bounds__(256) void mlp2_kernel(
    const float* __restrict__ hp, const float* __restrict__ w,
    const float* __restrict__ bias, float* __restrict__ out)
{
  const int wid = blockIdx.x * 8 + (threadIdx.x >> 5);
  const int lane = threadIdx.x & 31;
  const int o = wid & (cK * cD - 1);   // 16384
  const int b = wid >> 14;
  float s = 0.f;
#pragma unroll 4
  for (int j = 0; j < cHID / 32; ++j) {
    int kx = lane + 32 * j;
    s += hp[b * cHID + kx] * w[(size_t)kx * (cK * cD) + o];
  }
  s = wave_reduce_sum(s);
  if (lane == 0) out[(size_t)b * cK * cD + o] = s + bias[o];
}

// ---------------------------------------------------------------------------
// Embedding gather: one block per (b,s) row
// ---------------------------------------------------------------------------
__global__ __launch_bounds__(256) void embed_kernel(
    const int* __restrict__ ids, const float* __restrict__ emb, float* __restrict__ h)
{
  const int row = blockIdx.x, t = threadIdx.x;
  const int id = ids[row];
#pragma unroll
  for (int j = 0; j < 8; ++j)
    h[(size_t)row * cD + t + 256 * j] = emb[(size_t)id * cD + t + 256 * j];
}

// Scatter spectral tokens: one block per (b, k)
__global__ __launch_bounds__(256) void scatter_kernel(
    const int* __restrict__ spt, const float* __restrict__ spec, float* __restrict__ h)
{
  const int b = blockIdx.x >> 3, jk = blockIdx.x & 7, t = threadIdx.x;
  const int pos = spt[b] + jk;
#pragma unroll
  for (int j = 0; j < 8; ++j)
    h[((size_t)(b * cS + pos)) * cD + t + 256 * j] =
        spec[((size_t)(b * cK + jk)) * cD + t + 256 * j];
}

// ---------------------------------------------------------------------------
// RoPE in-place on (B,S,H,HD); thread per (row, freq-pair)
// ---------------------------------------------------------------------------
__global__ __launch_bounds__(256) void rope_kernel(float* __restrict__ x, int H)
{
  const int p   = blockIdx.x * 256 + threadIdx.x;
  const int i   = p & 63;            // freq index, HD/2 = 64
  const int row = p >> 6;            // (b*S+s)*H + h
  const int s_  = (row / H) & (cS - 1);
  const size_t base = (size_t)row * cHD + 2 * i;
  const float inv = __powf(10000.f, -(2.f * (float)i) / (float)cHD);
  const float ang = (float)s_ * inv;
  float sn, cs; __sincosf(ang, &sn, &cs);
  const float x0 = x[base], x1 = x[base + 1];
  x[base]     = x0 * cs - x1 * sn;
  x[base + 1] = x0 * sn + x1 * cs;
}

// ---------------------------------------------------------------------------
// Causal GQA attention, online softmax: one wave32 per (b,h,q) row.
// Lanes split HD=128 four-wide; cross-lane dot via shfl_xor tree.
// KV per layer (2 x 16 MB) lives in the 192 MB L2, so streaming re-reads are cheap.
// ---------------------------------------------------------------------------
__global__ __launch_bounds__(256) void attn_kernel(
    const float* __restrict__ q, const float* __restrict__ k,
    const float* __restrict__ v, float* __restrict__ o)
{
  const int wid  = blockIdx.x * 8 + (threadIdx.x >> 5);
  const int lane = threadIdx.x & 31;
  const int qi = wid & (cS - 1);
  const int h  = (wid >> 10) & (cNH - 1);
  const int b  = wid >> 14;
  const int kvh = h / (cNH / cNKV);
  const float scale = 0.08838834764831845f;   // 1/sqrt(128)

  const float4 qd = *(const float4*)&q[(((size_t)(b * cS + qi)) * cNH + h) * cHD + lane * 4];
  const float qx = qd.x * scale, qy = qd.y * scale, qz = qd.z * scale, qw = qd.w * scale;

  const float* kb = k + ((size_t)b * cS * cNKV + kvh) * cHD + lane * 4;
  const float* vb = v + ((size_t)b * cS * cNKV + kvh) * cHD + lane * 4;

  float m = -3.0e38f, l = 0.f;
  float ax = 0.f, ay = 0.f, az = 0.f, aw = 0.f;
  for (int j = 0; j <= qi; ++j) {
    const float4 kd = *(const float4*)(kb + (size_t)j * cNKV * cHD);
    float p = qx * kd.x + qy * kd.y + qz * kd.z + qw * kd.w;
    p = wave_reduce_sum(p);
    const float mn = fmaxf(m, p);
    const float eo = __expf(m - mn);
    const float ep = __expf(p - mn);
    l = l * eo + ep;
    const float4 vd = *(const float4*)(vb + (size_t)j * cNKV * cHD);
    ax = ax * eo + ep * vd.x; ay = ay * eo + ep * vd.y;
    az = az * eo + ep * vd.z; aw = aw * eo + ep * vd.w;
    m = mn;
  }
  const float inv = 1.f / l;
  float4 outv; outv.x = ax * inv; outv.y = ay * inv; outv.z = az * inv; outv.w = aw * inv;
  *(float4*)&o[(((size_t)(b * cS + qi)) * cNH + h) * cHD + lane * 4] = outv;
}

// SwiGLU gate: a1 = silu(a1) * a3
__global__ __launch_bounds__(256) void silumul_kernel(
    float* __restrict__ a1, const float* __restrict__ a3, size_t n)
{
  size_t i = (size_t)blockIdx.x * 256 + threadIdx.x;
  if (i < n) {
    float a = a1[i];
    a1[i] = (a / (1.f + __expf(-a))) * a3[i];
  }
}

// ---------------------------------------------------------------------------
extern "C" void kernel_launch(void* const* d_in, const int* in_sizes, int n_in,
                              void* d_out, int out_size, void* d_ws, size_t ws_size,
                              hipStream_t stream)
{
  const int*   input_ids = (const int*)  d_in[0];
  const float* spectra   = (const float*)d_in[1];
  const int*   spt       = (const int*)  d_in[2];
  const float* proj_w1   = (const float*)d_in[3];
  const float* proj_b1   = (const float*)d_in[4];
  const float* proj_w2   = (const float*)d_in[5];
  const float* proj_b2   = (const float*)d_in[6];
  const float* ln_g      = (const float*)d_in[7];
  const float* ln_b      = (const float*)d_in[8];
  const float* tok_emb   = (const float*)d_in[9];
  const float* attn_nw   = (const float*)d_in[10];
  const float* wq        = (const float*)d_in[11];
  const float* wk        = (const float*)d_in[12];
  const float* wv        = (const float*)d_in[13];
  const float* wo        = (const float*)d_in[14];
  const float* ffn_nw    = (const float*)d_in[15];
  const float* w1        = (const float*)d_in[16];
  const float* w3        = (const float*)d_in[17];
  const float* w2        = (const float*)d_in[18];
  const float* norm_w    = (const float*)d_in[19];
  const float* out_w     = (const float*)d_in[20];

  float* ws = (float*)d_ws;
  size_t off = 0;
  auto alloc = [&](size_t n) { float* p = ws + off; off += n; return p; };
  float* h     = alloc((size_t)cB * cS * cD);
  float* xb    = alloc((size_t)cB * cS * cD);
  float* qb    = alloc((size_t)cB * cS * cNH  * cHD);
  float* kb    = alloc((size_t)cB * cS * cNKV * cHD);
  float* vb    = alloc((size_t)cB * cS * cNKV * cHD);
  float* ab    = alloc((size_t)cB * cS * cNH  * cHD);
  float* a1    = alloc((size_t)cB * cS * cFFN);
  float* a3    = alloc((size_t)cB * cS * cFFN);
  float* hproj = alloc((size_t)cB * cHID);
  float* spec  = alloc((size_t)cB * cK * cD);

  const int M = cB * cS;  // 4096 token rows

  // --- spectra projection path ---
  mlp1_kernel<<<cB * cHID / 8, 256, 0, stream>>>(spectra, proj_w1, proj_b1, hproj);
  mlp2_kernel<<<cB * cK * cD / 8, 256, 0, stream>>>(hproj, proj_w2, proj_b2, spec);
  layernorm_kernel<<<cB * cK, 256, 0, stream>>>(spec, ln_g, ln_b);

  // --- embeddings ---
  embed_kernel<<<M, 256, 0, stream>>>(input_ids, tok_emb, h);
  scatter_kernel<<<cB * cK, 256, 0, stream>>>(spt, spec, h);

  // --- transformer layers ---
  for (int l = 0; l < 2; ++l) {
    rmsnorm_kernel<<<M, 256, 0, stream>>>(h, attn_nw + (size_t)l * cD, xb);

    gemm_wmma_f32<<<dim3(cD / 64, M / 64), 256, 0, stream>>>(
        xb, wq + (size_t)l * cD * (cNH * cHD), qb, M, cNH * cHD, cD, 0);
    gemm_wmma_f32<<<dim3((cNKV * cHD) / 64, M / 64), 256, 0, stream>>>(
        xb, wk + (size_t)l * cD * (cNKV * cHD), kb, M, cNKV * cHD, cD, 0);
    gemm_wmma_f32<<<dim3((cNKV * cHD) / 64, M / 64), 256, 0, stream>>>(
        xb, wv + (size_t)l * cD * (cNKV * cHD), vb, M, cNKV * cHD, cD, 0);

    rope_kernel<<<(cB * cS * cNH  * (cHD / 2)) / 256, 256, 0, stream>>>(qb, cNH);
    rope_kernel<<<(cB * cS * cNKV * (cHD / 2)) / 256, 256, 0, stream>>>(kb, cNKV);

    attn_kernel<<<(cB * cNH * cS) / 8, 256, 0, stream>>>(qb, kb, vb, ab);

    gemm_wmma_f32<<<dim3(cD / 64, M / 64), 256, 0, stream>>>(
        ab, wo + (size_t)l * (cNH * cHD) * cD, h, M, cD, cNH * cHD, 1);

    rmsnorm_kernel<<<M, 256, 0, stream>>>(h, ffn_nw + (size_t)l * cD, xb);

    gemm_wmma_f32<<<dim3(cFFN / 64, M / 64), 256, 0, stream>>>(
        xb, w1 + (size_t)l * cD * cFFN, a1, M, cFFN, cD, 0);
    gemm_wmma_f32<<<dim3(cFFN / 64, M / 64), 256, 0, stream>>>(
        xb, w3 + (size_t)l * cD * cFFN, a3, M, cFFN, cD, 0);
    silumul_kernel<<<(unsigned)(((size_t)M * cFFN) / 256), 256, 0, stream>>>(
        a1, a3, (size_t)M * cFFN);
    gemm_wmma_f32<<<dim3(cD / 64, M / 64), 256, 0, stream>>>(
        a1, w2 + (size_t)l * cFFN * cD, h, M, cD, cFFN, 1);
  }

  // --- final norm + logits ---
  rmsnorm_kernel<<<M, 256, 0, stream>>>(h, norm_w, xb);
  gemm_wmma_f32<<<dim3(cV / 64, M / 64), 256, 0, stream>>>(
      xb, out_w, (float*)d_out, M, cV, cD, 0);
}